// MultiHeadSelfAttention_57037165691282
// MI455X (gfx1250) — compile-verified
//
#include <hip/hip_runtime.h>

// MHA forward for B=2,S=2048,D=768,H=12,HD=64 (fp32), gfx1250 (CDNA5, wave32).
// Matmuls via V_WMMA_F32_16X16X4_F32 (exact fp32; problem is memory-bound by
// the 402MB attn_weights write). GEMM tile staging uses the CDNA5 async
// global->LDS B128 copy path (ASYNCcnt) when the builtin is available.

#define Bb 2
#define Ss 2048
#define Dd 768
#define Hh 12
#define HD 64
#define LST 36  // LDS row stride (floats): 144B rows -> 16B-aligned float4 slots

typedef float v2f __attribute__((ext_vector_type(2)));
typedef float v8f __attribute__((ext_vector_type(8)));
typedef int v4i __attribute__((vector_size(16)));

static __device__ __forceinline__ v8f wmma4(v2f a, v2f b, v8f c) {
  // D(16x16,f32) = A(16x4,f32) x B(4x16,f32) + C
  return __builtin_amdgcn_wmma_f32_16x16x4_f32(false, a, false, b, (short)0, c,
                                               false, false);
}

#if __has_builtin(__builtin_amdgcn_global_load_async_to_lds_b128)
#define ASYNC_CP 1
static __device__ __forceinline__ void cp_b128(const float* g, float* l) {
  __builtin_amdgcn_global_load_async_to_lds_b128(
      (__attribute__((address_space(1))) v4i*)g,
      (__attribute__((address_space(3))) v4i*)l, 0, 0);
}
#else
#define ASYNC_CP 0
#endif

// Stage A tile (64 x 32 @ ld) and B tile (32 x 32 @ ld) into padded LDS.
// 256 threads: A = 2x float4/thread, B = 1x float4/thread, all coalesced.
static __device__ __forceinline__ void stage_tiles(const float* __restrict__ gA,
                                                   const float* __restrict__ gB,
                                                   int ld, float* ldsA,
                                                   float* ldsB, int t) {
#if ASYNC_CP
#pragma unroll
  for (int i = 0; i < 2; ++i) {
    int idx = i * 256 + t;            // 0..511
    int row = idx >> 3, c4 = (idx & 7) * 4;
    cp_b128(&gA[(size_t)row * ld + c4], &ldsA[row * LST + c4]);
  }
  {
    int row = t >> 3, c4 = (t & 7) * 4;
    cp_b128(&gB[(size_t)row * ld + c4], &ldsB[row * LST + c4]);
  }
  __asm__ volatile("s_wait_asynccnt 0" ::: "memory");
#else
  float4 ra0, ra1, rb;
  {
    int idx = t;
    ra0 = *(const float4*)&gA[(size_t)(idx >> 3) * ld + (idx & 7) * 4];
    idx = 256 + t;
    ra1 = *(const float4*)&gA[(size_t)(idx >> 3) * ld + (idx & 7) * 4];
    rb  = *(const float4*)&gB[(size_t)(t >> 3) * ld + (t & 7) * 4];
  }
  *(float4*)&ldsA[(t >> 3) * LST + (t & 7) * 4] = ra0;
  *(float4*)&ldsA[((256 + t) >> 3) * LST + (t & 7) * 4] = ra1;
  *(float4*)&ldsB[(t >> 3) * LST + (t & 7) * 4] = rb;
#endif
}

// ---------------------------------------------------------------------------
// Kernel 1: QKV projection. grid=(M/64, N/32, 3), block=256 (8 waves).
// z=0 -> Q [B,H,S,HD]; z=1 -> K^T [B,H,HD,S] (LDS transpose); z=2 -> V.
// ---------------------------------------------------------------------------
__global__ __launch_bounds__(256) void qkv_kernel(
    const float* __restrict__ x,
    const float* __restrict__ Wq, const float* __restrict__ bq,
    const float* __restrict__ Wk, const float* __restrict__ bk,
    const float* __restrict__ Wv, const float* __restrict__ bv,
    float* __restrict__ q_ws, float* __restrict__ kT_ws,
    float* __restrict__ v_ws) {
  __shared__ float ldsA[64 * LST];
  __shared__ float ldsB[32 * LST];
  const int t = threadIdx.x;
  const int wave = t >> 5, lane = t & 31;
  const int lrow = lane & 15, lhi = lane >> 4, lcol = lane & 15;
  const int mw = (wave >> 1) * 16, nw = (wave & 1) * 16;
  const int m0 = blockIdx.x * 64;
  const int n0 = blockIdx.y * 32;
  const int z = blockIdx.z;
  const float* W = (z == 0) ? Wq : ((z == 1) ? Wk : Wv);
  const float* bias = (z == 0) ? bq : ((z == 1) ? bk : bv);

  v8f acc = {};
  for (int k0 = 0; k0 < Dd; k0 += 32) {
    stage_tiles(&x[(size_t)m0 * Dd + k0], &W[(size_t)k0 * Dd + n0], Dd, ldsA,
                ldsB, t);
    __syncthreads();
#pragma unroll
    for (int s = 0; s < 8; ++s) {
      int kk = s * 4 + 2 * lhi;
      v2f a, b;
      a.x = ldsA[(mw + lrow) * LST + kk];
      a.y = ldsA[(mw + lrow) * LST + kk + 1];
      b.x = ldsB[kk * LST + nw + lcol];
      b.y = ldsB[(kk + 1) * LST + nw + lcol];
      acc = wmma4(a, b, acc);
    }
    __syncthreads();
  }
  const float bb_n = bias[n0 + nw + lcol];
#pragma unroll
  for (int r = 0; r < 8; ++r) acc[r] += bb_n;

  if (z == 1) {
    // Transpose 64(m) x 32(n) block tile via LDS, store K^T[b][h][d][s].
#pragma unroll
    for (int r = 0; r < 8; ++r) {
      int m = mw + r + 8 * lhi;
      ldsA[m * LST + nw + lcol] = acc[r];
    }
    __syncthreads();
#pragma unroll
    for (int i = 0; i < 2; ++i) {
      int idx = i * 256 + t;            // 0..511 float4 slots
      int m4 = (idx & 15) * 4, nloc = idx >> 4;
      int ng = n0 + nloc;
      int h = ng >> 6, d = ng & 63;
      int mg = m0 + m4;
      int bi = mg >> 11, s = mg & 2047;
      float4 vv;
      vv.x = ldsA[(m4 + 0) * LST + nloc];
      vv.y = ldsA[(m4 + 1) * LST + nloc];
      vv.z = ldsA[(m4 + 2) * LST + nloc];
      vv.w = ldsA[(m4 + 3) * LST + nloc];
      *(float4*)&kT_ws[((size_t)(bi * Hh + h) * HD + d) * Ss + s] = vv;
    }
  } else {
    float* dst = (z == 0) ? q_ws : v_ws;
    int ng = n0 + nw + lcol;
    int h = ng >> 6, d = ng & 63;
#pragma unroll
    for (int r = 0; r < 8; ++r) {
      int mg = m0 + mw + r + 8 * lhi;
      int bi = mg >> 11, s = mg & 2047;
      dst[((size_t)(bi * Hh + h) * Ss + s) * HD + d] = acc[r];
    }
  }
}

// ---------------------------------------------------------------------------
// Kernel 2: causal flash attention, one wave per 16-query tile.
// grid=(S/64, H, B), block=128 (4 waves). Writes normalized P to attnW once,
// accumulates O = P*V via WMMA (P transposed through per-wave LDS slab).
// ---------------------------------------------------------------------------
__global__ __launch_bounds__(128) void attn_kernel(
    const float* __restrict__ q_ws, const float* __restrict__ kT_ws,
    const float* __restrict__ v_ws, float* __restrict__ attnW,
    float* __restrict__ ao_ws) {
  __shared__ float ldsP[4][16 * 17];
  const int t = threadIdx.x;
  const int wave = t >> 5, lane = t & 31;
  const int lrow = lane & 15, lhi = lane >> 4, lcol = lane & 15;
  const int h = blockIdx.y, b = blockIdx.z;
  const int bh = b * Hh + h;
  const int q_base = blockIdx.x * 64 + wave * 16;
  const float* Q = q_ws + (size_t)bh * Ss * HD;
  const float* KT = kT_ws + (size_t)bh * HD * Ss;
  const float* V = v_ws + (size_t)bh * Ss * HD;
  float* attn = attnW + (size_t)bh * Ss * Ss;
  float* ldsp = ldsP[wave];

  // Q tile (16x64) in registers, pre-scaled by 1/sqrt(HD) = 0.125.
  v2f qa[16];
#pragma unroll
  for (int s = 0; s < 16; ++s) {
    const float* p = &Q[(size_t)(q_base + lrow) * HD + s * 4 + 2 * lhi];
    qa[s].x = p[0] * 0.125f;
    qa[s].y = p[1] * 0.125f;
  }

  const int tlim = q_base >> 4;  // last key tile touching the causal region
  float rm[8], rl[8];
#pragma unroll
  for (int r = 0; r < 8; ++r) { rm[r] = -3.0e38f; rl[r] = 0.0f; }

  // ---- Pass 1: online row max / exp-sum over causal tiles ----
  for (int tt = 0; tt <= tlim; ++tt) {
    const int kb = tt * 16;
    v8f acc = {};
#pragma unroll
    for (int s = 0; s < 16; ++s) {
      int d = s * 4 + 2 * lhi;
      v2f bf;
      bf.x = KT[(size_t)d * Ss + kb + lcol];
      bf.y = KT[(size_t)(d + 1) * Ss + kb + lcol];
      acc = wmma4(qa[s], bf, acc);
    }
    if (tt == tlim) {
#pragma unroll
      for (int r = 0; r < 8; ++r) {
        int m = r + 8 * lhi;
        if (lcol > m) acc[r] = -3.0e38f;  // causal mask (key > query)
      }
    }
#pragma unroll
    for (int r = 0; r < 8; ++r) {
      float tmax = acc[r];
      tmax = fmaxf(tmax, __shfl_xor(tmax, 1));
      tmax = fmaxf(tmax, __shfl_xor(tmax, 2));
      tmax = fmaxf(tmax, __shfl_xor(tmax, 4));
      tmax = fmaxf(tmax, __shfl_xor(tmax, 8));
      float newm = fmaxf(rm[r], tmax);
      float pe = __expf(acc[r] - newm);
      float tsum = pe;
      tsum += __shfl_xor(tsum, 1);
      tsum += __shfl_xor(tsum, 2);
      tsum += __shfl_xor(tsum, 4);
      tsum += __shfl_xor(tsum, 8);
      rl[r] = rl[r] * __expf(rm[r] - newm) + tsum;
      rm[r] = newm;
    }
  }

  float linv[8];
#pragma unroll
  for (int r = 0; r < 8; ++r) linv[r] = 1.0f / rl[r];

  // ---- Pass 2: recompute scores, emit normalized P, accumulate O = P*V ----
  v8f oacc[4] = {};
  for (int tt = 0; tt < Ss / 16; ++tt) {
    const int kb = tt * 16;
    if (tt <= tlim) {
      v8f acc = {};
#pragma unroll
      for (int s = 0; s < 16; ++s) {
        int d = s * 4 + 2 * lhi;
        v2f bf;
        bf.x = KT[(size_t)d * Ss + kb + lcol];
        bf.y = KT[(size_t)(d + 1) * Ss + kb + lcol];
        acc = wmma4(qa[s], bf, acc);
      }
#pragma unroll
      for (int r = 0; r < 8; ++r) {
        int m = r + 8 * lhi;
        float sv = acc[r];
        if (tt == tlim && lcol > m) sv = -3.0e38f;
        float pv = __expf(sv - rm[r]) * linv[r];
        attn[(size_t)(q_base + m) * Ss + kb + lcol] = pv;
        ldsp[m * 17 + lcol] = pv;  // stage for C-layout -> A-layout transpose
      }
      __asm__ volatile("s_wait_dscnt 0" ::: "memory");
      v2f pa[4];
#pragma unroll
      for (int s = 0; s < 4; ++s) {
        int kk = s * 4 + 2 * lhi;
        pa[s].x = ldsp[lrow * 17 + kk];
        pa[s].y = ldsp[lrow * 17 + kk + 1];
      }
      __asm__ volatile("" ::: "memory");  // keep reads before next tile writes
#pragma unroll
      for (int nt = 0; nt < 4; ++nt) {
#pragma unroll
        for (int s = 0; s < 4; ++s) {
          int kk = s * 4 + 2 * lhi;
          v2f bf;
          bf.x = V[(size_t)(kb + kk) * HD + nt * 16 + lcol];
          bf.y = V[(size_t)(kb + kk + 1) * HD + nt * 16 + lcol];
          oacc[nt] = wmma4(pa[s], bf, oacc[nt]);
        }
      }
    } else {
      // fully-masked tiles: attention weights are exactly zero
#pragma unroll
      for (int r = 0; r < 8; ++r)
        attn[(size_t)(q_base + r + 8 * lhi) * Ss + kb + lcol] = 0.0f;
    }
  }

  // store O in [B,S,D] layout (heads re-interleaved) for the out-projection
#pragma unroll
  for (int nt = 0; nt < 4; ++nt) {
#pragma unroll
    for (int r = 0; r < 8; ++r) {
      int qg = q_base + r + 8 * lhi;
      ao_ws[(size_t)(b * Ss + qg) * Dd + h * HD + nt * 16 + lcol] = oacc[nt][r];
    }
  }
}

// ---------------------------------------------------------------------------
// Kernel 3: output projection out = AO @ Wo + bo. grid=(M/64, N/32), block=256.
// ---------------------------------------------------------------------------
__global__ __launch_bounds__(256) void proj_kernel(
    const float* __restrict__ A, const float* __restrict__ W,
    const float* __restrict__ bias, float* __restrict__ out) {
  __shared__ float ldsA[64 * LST];
  __shared__ float ldsB[32 * LST];
  const int t = threadIdx.x;
  const int wave = t >> 5, lane = t & 31;
  const int lrow = lane & 15, lhi = lane >> 4, lcol = lane & 15;
  const int mw = (wave >> 1) * 16, nw = (wave & 1) * 16;
  const int m0 = blockIdx.x * 64;
  const int n0 = blockIdx.y * 32;

  v8f acc = {};
  for (int k0 = 0; k0 < Dd; k0 += 32) {
    stage_tiles(&A[(size_t)m0 * Dd + k0], &W[(size_t)k0 * Dd + n0], Dd, ldsA,
                ldsB, t);
    __syncthreads();
#pragma unroll
    for (int s = 0; s < 8; ++s) {
      int kk = s * 4 + 2 * lhi;
      v2f a, b;
      a.x = ldsA[(mw + lrow) * LST + kk];
      a.y = ldsA[(mw + lrow) * LST + kk + 1];
      b.x = ldsB[kk * LST + nw + lcol];
      b.y = ldsB[(kk + 1) * LST + nw + lcol];
      acc = wmma4(a, b, acc);
    }
    __syncthreads();
  }
  const float bb_n = bias[n0 + nw + lcol];
#pragma unroll
  for (int r = 0; r < 8; ++r) {
    int mg = m0 + mw + r + 8 * lhi;
    out[(size_t)mg * Dd + n0 + nw + lcol] = acc[r] + bb_n;
  }
}

// ---------------------------------------------------------------------------
extern "C" void kernel_launch(void* const* d_in, const int* in_sizes, int n_in,
                              void* d_out, int out_size, void* d_ws,
                              size_t ws_size, hipStream_t stream) {
  const float* x  = (const float*)d_in[0];
  // d_in[1] = mask (int32 tril) — causality computed analytically
  const float* Wq = (const float*)d_in[2];
  const float* bq = (const float*)d_in[3];
  const float* Wk = (const float*)d_in[4];
  const float* bk = (const float*)d_in[5];
  const float* Wv = (const float*)d_in[6];
  const float* bv = (const float*)d_in[7];
  const float* Wo = (const float*)d_in[8];
  const float* bo = (const float*)d_in[9];

  float* out = (float*)d_out;                       // [B,S,D]
  float* attnW = out + (size_t)Bb * Ss * Dd;        // [B,H,S,S]

  float* ws = (float*)d_ws;
  const size_t per = (size_t)Bb * Hh * Ss * HD;     // 3,145,728 floats
  float* q_ws  = ws;                                // [B,H,S,HD]
  float* kT_ws = ws + per;                          // [B,H,HD,S]
  float* v_ws  = ws + 2 * per;                      // [B,H,S,HD]
  float* ao_ws = ws + 3 * per;                      // [B,S,D]

  dim3 g1((Bb * Ss) / 64, Dd / 32, 3);
  qkv_kernel<<<g1, 256, 0, stream>>>(x, Wq, bq, Wk, bk, Wv, bv, q_ws, kT_ws,
                                     v_ws);
  dim3 g2(Ss / 64, Hh, Bb);
  attn_kernel<<<g2, 128, 0, stream>>>(q_ws, kT_ws, v_ws, attnW, ao_ws);
  dim3 g3((Bb * Ss) / 64, Dd / 32, 1);
  proj_kernel<<<g3, 256, 0, stream>>>(ao_ws, Wo, bo, out);
}